// GNNModel_33320356283137
// MI455X (gfx1250) — compile-verified
//
#include <hip/hip_runtime.h>
#include <hip/hip_bf16.h>

typedef __attribute__((ext_vector_type(2))) float v2f;
typedef __attribute__((ext_vector_type(8))) float v8f;

// ---------------------------------------------------------------------------
// Hardware fp32 atomic add (global_atomic_add_f32, no return) — the scatter
// phase is the bottleneck and must not fall back to a CAS loop.
// ---------------------------------------------------------------------------
__device__ __forceinline__ void atomAddF(float* p, float v) {
    unsafeAtomicAdd(p, v);
}

// ---------------------------------------------------------------------------
// zero fill
// ---------------------------------------------------------------------------
__global__ void k_zero(float* __restrict__ p, int n) {
    int i = blockIdx.x * blockDim.x + threadIdx.x;
    if (i < n) p[i] = 0.0f;
}

// ---------------------------------------------------------------------------
// degree count: deg[col[e]] += 1  (self-loop added later as +1 in rsqrt)
// ---------------------------------------------------------------------------
__global__ void k_degree(const int* __restrict__ cols, float* __restrict__ deg, int E) {
    int i = blockIdx.x * blockDim.x + threadIdx.x;
    if (i < E) atomAddF(&deg[cols[i]], 1.0f);
}

// dinv[i] = rsqrt(deg[i] + 1)   (deg>=1 always due to self loop)
__global__ void k_rsqrt(float* __restrict__ d, int n) {
    int i = blockIdx.x * blockDim.x + threadIdx.x;
    if (i < n) d[i] = rsqrtf(d[i] + 1.0f);
}

// ---------------------------------------------------------------------------
// WMMA fp32 GEMM:  C[nRows x DO] = H[nRows x K] * W[DO x K]^T
// One wave per 16x16 output tile; K consumed 4 at a time with
// V_WMMA_F32_16X16X4_F32. nRows, DO multiples of 16; K multiple of 4.
//
// A 16x4 layout: lane l, half=l>>4, m=l&15:  a = {H[m0+m, k0+2*half + {0,1}]}
// B  4x16 layout: lane l:                    b = {W[o0+m, k0+2*half + {0,1}]}
// C 16x16 layout: VGPR j -> row m0+8*half+j, col o0+m
// ---------------------------------------------------------------------------
__global__ void k_gemm_wmma(const float* __restrict__ H,
                            const float* __restrict__ W,
                            float* __restrict__ C,
                            int nRows, int K, int DO) {
    const int lane   = threadIdx.x & 31;
    const int waveId = threadIdx.x >> 5;
    const int colTiles = DO >> 4;
    const int numTiles = (nRows >> 4) * colTiles;

    int tile = blockIdx.x * (blockDim.x >> 5) + waveId;   // wave-uniform
    if (tile >= numTiles) return;                          // whole wave exits: EXEC stays all-1s for WMMA

    const int rowTile = tile / colTiles;
    const int colTile = tile - rowTile * colTiles;
    const int m0 = rowTile << 4;
    const int o0 = colTile << 4;

    const int half = lane >> 4;      // 0: K lo pair, 1: K hi pair
    const int m    = lane & 15;

    const float* __restrict__ hrow = H + (size_t)(m0 + m) * K + 2 * half;
    const float* __restrict__ wrow = W + (size_t)(o0 + m) * K + 2 * half;

    v8f acc = {};
#pragma unroll 8
    for (int k0 = 0; k0 < K; k0 += 4) {
        v2f a, b;
        a.x = hrow[k0];  a.y = hrow[k0 + 1];
        b.x = wrow[k0];  b.y = wrow[k0 + 1];
        acc = __builtin_amdgcn_wmma_f32_16x16x4_f32(
            /*neg_a=*/false, a, /*neg_b=*/false, b,
            /*c_mod=*/(short)0, acc, /*reuse_a=*/false, /*reuse_b=*/false);
    }

    const int r0 = m0 + 8 * half;
#pragma unroll
    for (int j = 0; j < 8; ++j) {
        C[(size_t)(r0 + j) * DO + o0 + m] = acc[j];
    }
}

// ---------------------------------------------------------------------------
// Edge scatter: agg[col[e], :] += tmp[row[e], :] * dinv[row]*dinv[col]
// One wave per edge: 32 lanes x (DO/32) contiguous floats = fully coalesced
// gather; scatter via hardware fp32 atomics (L2-resident: 25.6MB << 192MB L2).
// ---------------------------------------------------------------------------
__global__ void k_edge_scatter(const float* __restrict__ tmp,
                               const int* __restrict__ rows,
                               const int* __restrict__ cols,
                               const float* __restrict__ dinv,
                               float* __restrict__ agg,
                               int E, int DO) {
    const int gid  = blockIdx.x * blockDim.x + threadIdx.x;
    const int e    = gid >> 5;
    const int lane = threadIdx.x & 31;
    if (e >= E) return;

    const int r = rows[e];
    const int c = cols[e];
    const float norm = dinv[r] * dinv[c];

    const int per = DO >> 5;                         // 4 (DO=128) or 2 (DO=64)
    const float* __restrict__ src = tmp + (size_t)r * DO + lane * per;
    float*                    dst = agg + (size_t)c * DO + lane * per;

#pragma unroll 4
    for (int i = 0; i < per; ++i)
        atomAddF(&dst[i], src[i] * norm);
}

// ---------------------------------------------------------------------------
// Finish: out = [relu]( agg + tmp * dinv^2 (self loop) + bias )
// May run in-place (out == agg): each thread touches index i only.
// ---------------------------------------------------------------------------
__global__ void k_finish(const float* agg,
                         const float* __restrict__ tmp,
                         const float* __restrict__ dinv,
                         const float* __restrict__ bias,
                         float* out,
                         int total, int doShift, int doRelu) {
    int i = blockIdx.x * blockDim.x + threadIdx.x;
    if (i >= total) return;
    const int DOmask = (1 << doShift) - 1;
    const int node = i >> doShift;
    const int f    = i & DOmask;
    const float d  = dinv[node];
    float v = agg[i] + tmp[i] * d * d + bias[f];
    out[i] = doRelu ? fmaxf(v, 0.0f) : v;
}

// ---------------------------------------------------------------------------
// Host orchestration
// ---------------------------------------------------------------------------
static inline int cdiv(long long a, long long b) { return (int)((a + b - 1) / b); }

extern "C" void kernel_launch(void* const* d_in, const int* in_sizes, int n_in,
                              void* d_out, int out_size, void* d_ws, size_t ws_size,
                              hipStream_t stream) {
    const float* x    = (const float*)d_in[0];
    const int*   eidx = (const int*)  d_in[1];
    const float* W0 = (const float*)d_in[2]; const float* b0 = (const float*)d_in[3];
    const float* W1 = (const float*)d_in[4]; const float* b1 = (const float*)d_in[5];
    const float* W2 = (const float*)d_in[6]; const float* b2 = (const float*)d_in[7];
    const float* W3 = (const float*)d_in[8]; const float* b3 = (const float*)d_in[9];

    const int D_IN = 128, D_H = 128, D_OUT = 64;
    const int N = in_sizes[0] / D_IN;         // 50000 (multiple of 16)
    const int E = in_sizes[1] / 2;            // 600000
    const int* rows = eidx;                   // edge_index[0] = sources
    const int* cols = eidx + E;               // edge_index[1] = targets

    // workspace: three rotating N x 128 fp32 buffers + dinv[N]
    float* A    = (float*)d_ws;
    float* B    = A + (size_t)N * D_H;
    float* C    = B + (size_t)N * D_H;
    float* dinv = C + (size_t)N * D_H;

    const int T = 256;

    // --- normalization coefficients: dinv = rsqrt(in_degree + 1) ------------
    k_zero  <<<cdiv(N, T), T, 0, stream>>>(dinv, N);
    k_degree<<<cdiv(E, T), T, 0, stream>>>(cols, dinv, E);
    k_rsqrt <<<cdiv(N, T), T, 0, stream>>>(dinv, N);

    // one GCN layer: in --gemm--> tmp; agg = scatter(tmp); out = finish(...)
    auto layer = [&](const float* in, const float* W, const float* b,
                     float* tmp, float* agg, float* out,
                     int DO, int doShift, int relu) {
        const int tiles = (N >> 4) * (DO >> 4);
        k_gemm_wmma<<<cdiv(tiles, T / 32), T, 0, stream>>>(in, W, tmp, N, D_H, DO);
        k_zero<<<cdiv((long long)N * DO, T), T, 0, stream>>>(agg, N * DO);
        k_edge_scatter<<<cdiv((long long)E * 32, T), T, 0, stream>>>(tmp, rows, cols, dinv, agg, E, DO);
        k_finish<<<cdiv((long long)N * DO, T), T, 0, stream>>>(agg, tmp, dinv, b, out, N * DO, doShift, relu);
    };

    // L0: x -> B      (tmp=A, agg=B, finish in place)
    layer(x, W0, b0, A, B, B, D_H, 7, 1);
    // L1: B -> C
    layer(B, W1, b1, A, C, C, D_H, 7, 1);
    // L2: C -> B
    layer(C, W2, b2, A, B, B, D_H, 7, 1);
    // L3: B -> d_out  (DO=64, no relu)
    layer(B, W3, b3, A, C, (float*)d_out, D_OUT, 6, 0);

    (void)n_in; (void)out_size; (void)ws_size;
}